// CRD_5789615915289
// MI455X (gfx1250) — compile-verified
//
#include <hip/hip_runtime.h>
#include <hip/hip_bf16.h>

// ---------------------------------------------------------------------------
// GCN layer for MI455X (gfx1250, wave32):
//   xw = x @ W            -> V_WMMA_F32_16X16X4_F32 (f32 WMMA, K-loop of 32)
//   deg/norm              -> atomic degree count + v_rsq
//   scatter aggregation   -> coalesced global_atomic_add_f32 into d_out
//   epilogue (in place)   -> self-loop term + bias + relu + threefry dropout
// ---------------------------------------------------------------------------

typedef __attribute__((ext_vector_type(2))) float v2f;
typedef __attribute__((ext_vector_type(8))) float v8f;

#define D_IN  128
#define D_OUT 64

// ---------------------- degree / normalization -----------------------------

// deg[i] = 1.0 (self loop contribution), executed before edge accumulation
__global__ void CRD_deg_init(float* __restrict__ deg, int N) {
    int i = blockIdx.x * blockDim.x + threadIdx.x;
    if (i < N) deg[i] = 1.0f;
}

__global__ void CRD_deg_edges(const int* __restrict__ dst, float* __restrict__ deg, int E) {
    int e = blockIdx.x * blockDim.x + threadIdx.x;
    if (e < E) atomicAdd(&deg[dst[e]], 1.0f);
}

// in-place: deg[i] = rsqrt(deg[i])   (deg >= 1 always due to self loops)
__global__ void CRD_rsq(float* __restrict__ deg, int N) {
    int i = blockIdx.x * blockDim.x + threadIdx.x;
    if (i < N) deg[i] = rsqrtf(deg[i]);
}

// zero the accumulation target (d_out)
__global__ void CRD_zero(float* __restrict__ p, size_t n) {
    size_t i = (size_t)blockIdx.x * blockDim.x + threadIdx.x;
    if (i < n) p[i] = 0.0f;
}

// ---------------------------- WMMA GEMM ------------------------------------
// xw[N,64] = x[N,128] @ W[128,64]; one wave computes one 16x16 output tile.
__global__ __launch_bounds__(256) void CRD_gemm_xw(const float* __restrict__ x,
                                                   const float* __restrict__ W,
                                                   float* __restrict__ xw,
                                                   int N) {
    const int wavesPerBlock = 256 / 32;
    int wave = blockIdx.x * wavesPerBlock + (int)(threadIdx.x >> 5);
    int lane = (int)(threadIdx.x & 31);

    const int colTiles   = D_OUT / 16;                 // 4
    const int totalWaves = (N / 16) * colTiles;        // N % 16 == 0 for this problem
    if (wave >= totalWaves) return;                    // wave-uniform exit (EXEC stays all-1s)

    int rowTile = wave / colTiles;
    int colTile = wave % colTiles;
    int row0 = rowTile * 16;
    int col0 = colTile * 16;

    int m  = lane & 15;       // row within tile (A) / column within tile (B,N index)
    int hi = lane >> 4;       // 0: K offset 0, 1: K offset 2 (A/B half-lane split)

    v8f acc = {};
    const float* arow = x + (size_t)(row0 + m) * D_IN;

    #pragma unroll 4
    for (int k = 0; k < D_IN; k += 4) {
        // A 16x4 f32: VGPR0 = K=(k+2*hi), VGPR1 = K=(k+2*hi+1), row = m
        v2f a;
        a.x = arow[k + 2 * hi + 0];
        a.y = arow[k + 2 * hi + 1];
        // B 4x16 f32: VGPR0 = row K=(k+2*hi), VGPR1 = row K=(k+2*hi+1), col = m
        const float* bp = W + (size_t)(k + 2 * hi) * D_OUT + col0 + m;
        v2f b;
        b.x = bp[0];
        b.y = bp[D_OUT];
        acc = __builtin_amdgcn_wmma_f32_16x16x4_f32(
            /*neg_a=*/false, a, /*neg_b=*/false, b,
            /*c_mod=*/(short)0, acc, /*reuse_a=*/false, /*reuse_b=*/false);
    }

    // D 16x16 f32 layout: VGPR i -> (M=i, lanes 0-15) / (M=i+8, lanes 16-31), N=lane&15
    #pragma unroll
    for (int i = 0; i < 8; ++i) {
        xw[(size_t)(row0 + i + 8 * hi) * D_OUT + col0 + m] = acc[i];
    }
}

// --------------------------- edge scatter ----------------------------------
// t = edge*64 + j ; wave handles a contiguous 32-float slice of one edge ->
// wave-uniform src/dst/dinv, coalesced gather of xw[src] and coalesced atomics.
__global__ __launch_bounds__(256) void CRD_scatter(const int* __restrict__ src,
                                                   const int* __restrict__ dst,
                                                   const float* __restrict__ dinv,
                                                   const float* __restrict__ xw,
                                                   float* __restrict__ out,
                                                   int E) {
    size_t t = (size_t)blockIdx.x * blockDim.x + threadIdx.x;
    int e = (int)(t >> 6);
    int j = (int)(t & 63);
    if (e >= E) return;
    int s = src[e];
    int d = dst[e];
    float w = dinv[s] * dinv[d];
    atomicAdd(out + (size_t)d * D_OUT + j, w * xw[(size_t)s * D_OUT + j]);
}

// ----------------------- threefry2x32 (JAX PRNG) ---------------------------

__device__ __forceinline__ unsigned CRD_rotl(unsigned x, int n) {
    return (x << n) | (x >> (32 - n));
}

__device__ __forceinline__ void CRD_threefry2x32(unsigned k0, unsigned k1,
                                                 unsigned c0, unsigned c1,
                                                 unsigned& o0, unsigned& o1) {
    const unsigned ks2 = k0 ^ k1 ^ 0x1BD11BDAu;
    unsigned x0 = c0 + k0;
    unsigned x1 = c1 + k1;
    const int ra[4] = {13, 15, 26, 6};
    const int rb[4] = {17, 29, 16, 24};
    #pragma unroll
    for (int i = 0; i < 4; ++i) { x0 += x1; x1 = CRD_rotl(x1, ra[i]); x1 ^= x0; }
    x0 += k1;  x1 += ks2 + 1u;
    #pragma unroll
    for (int i = 0; i < 4; ++i) { x0 += x1; x1 = CRD_rotl(x1, rb[i]); x1 ^= x0; }
    x0 += ks2; x1 += k0 + 2u;
    #pragma unroll
    for (int i = 0; i < 4; ++i) { x0 += x1; x1 = CRD_rotl(x1, ra[i]); x1 ^= x0; }
    x0 += k0;  x1 += k1 + 3u;
    #pragma unroll
    for (int i = 0; i < 4; ++i) { x0 += x1; x1 = CRD_rotl(x1, rb[i]); x1 ^= x0; }
    x0 += k1;  x1 += ks2 + 4u;
    #pragma unroll
    for (int i = 0; i < 4; ++i) { x0 += x1; x1 = CRD_rotl(x1, ra[i]); x1 ^= x0; }
    x0 += ks2; x1 += k0 + 5u;
    o0 = x0; o1 = x1;
}

// ------------------------------ epilogue -----------------------------------
// out[i,j] = dropout(relu(out[i,j] + dinv[i]^2 * xw[i,j] + b[j]))
__global__ __launch_bounds__(256) void CRD_epilogue(const float* __restrict__ xw,
                                                    const float* __restrict__ dinv,
                                                    const float* __restrict__ b,
                                                    float* __restrict__ out,
                                                    int N) {
    size_t idx = (size_t)blockIdx.x * blockDim.x + threadIdx.x;
    size_t total = (size_t)N * D_OUT;
    if (idx >= total) return;
    int i = (int)(idx >> 6);
    int j = (int)(idx & 63);

    float di = dinv[i];
    float v = out[idx] + di * di * xw[idx] + b[j];
    v = fmaxf(v, 0.0f);

    // JAX: keep = uniform(key(42), shape) < 0.5, threefry2x32 counter mode.
    // Flat counter iota is split in half: idx < half -> x0 of (idx, idx+half),
    // else x1 of (idx-half, idx). key(42) -> (k0,k1) = (0,42).
    const unsigned k0 = 0u, k1 = 42u;
    unsigned half = (unsigned)(total >> 1);
    unsigned id = (unsigned)idx;
    bool lohalf = id < half;
    unsigned c0 = lohalf ? id : (id - half);
    unsigned c1 = lohalf ? (id + half) : id;
    unsigned o0, o1;
    CRD_threefry2x32(k0, k1, c0, c1, o0, o1);
    unsigned bits = lohalf ? o0 : o1;
    float u = __uint_as_float((bits >> 9) | 0x3F800000u) - 1.0f;

    out[idx] = (u < 0.5f) ? v * 2.0f : 0.0f;   // /(1-p) with p=0.5
}

// ------------------------------- launch ------------------------------------

extern "C" void kernel_launch(void* const* d_in, const int* in_sizes, int n_in,
                              void* d_out, int out_size, void* d_ws, size_t ws_size,
                              hipStream_t stream) {
    const float* x  = (const float*)d_in[0];
    const int*   ei = (const int*)d_in[1];
    const float* W  = (const float*)d_in[2];
    const float* b  = (const float*)d_in[3];
    float* out = (float*)d_out;

    const int N = in_sizes[0] / D_IN;     // 100000
    const int E = in_sizes[1] / 2;        // 1600000
    const int* src = ei;
    const int* dst = ei + E;

    // workspace layout: deg/dinv [N] | xw [N*64]
    char* ws = (char*)d_ws;
    float* deg = (float*)ws;                                   // N floats (becomes dinv)
    size_t degBytes = ((size_t)N * sizeof(float) + 255) & ~(size_t)255;
    float* xw = (float*)(ws + degBytes);                       // N*64 floats

    const int T = 256;
    size_t outElems = (size_t)N * D_OUT;

    // 1) zero accumulation target, init degrees with self-loops
    CRD_zero<<<(int)((outElems + T - 1) / T), T, 0, stream>>>(out, outElems);
    CRD_deg_init<<<(N + T - 1) / T, T, 0, stream>>>(deg, N);

    // 2) degree accumulation over edges
    CRD_deg_edges<<<(E + T - 1) / T, T, 0, stream>>>(dst, deg, E);

    // 3) dinv = rsqrt(deg), in place
    CRD_rsq<<<(N + T - 1) / T, T, 0, stream>>>(deg, N);

    // 4) xw = x @ W via f32 WMMA (one 16x16 tile per wave, 8 waves per block)
    {
        int totalWaves = (N / 16) * (D_OUT / 16);
        int blocks = (totalWaves + 7) / 8;
        CRD_gemm_xw<<<blocks, T, 0, stream>>>(x, W, xw, N);
    }

    // 5) edge scatter: out[dst] += dinv[src]*dinv[dst] * xw[src]
    {
        size_t threads = (size_t)E * D_OUT;
        CRD_scatter<<<(int)((threads + T - 1) / T), T, 0, stream>>>(src, dst, deg, xw, out, E);
    }

    // 6) self-loop term + bias + relu + deterministic threefry dropout, in place
    CRD_epilogue<<<(int)((outElems + T - 1) / T), T, 0, stream>>>(xw, deg, b, out, N);
}